// DendriticNeuron_72567767433458
// MI455X (gfx1250) — compile-verified
//
#include <hip/hip_runtime.h>

typedef float v2f __attribute__((ext_vector_type(2)));
typedef float v4f __attribute__((ext_vector_type(4)));
typedef float v8f __attribute__((ext_vector_type(8)));

#define BATCH        1024
#define N_NEURONS    8192
#define N_BRANCHES   8
#define IPB          64
#define TOTAL_INPUTS 512

// LDS row stride (floats): 512 + 4 so row-strided fragment reads hit distinct banks
#define XS_STRIDE    516

// precomputed decay constants
#define SYN_DECAY      0.99335551f   // exp(-0.1/15)
#define PLATEAU_DECAY  0.99875078f   // exp(-0.1/80)
#define E_DECAY        0.98019867f   // exp(-0.1/5)
#define DT_OVER_TAU    0.005f        // 0.1/20
#define LOG2E          1.44269504f

// ---------------------------------------------------------------------------
// Prep pass: clamp weights at 0 and swizzle into WMMA B-fragment order:
//   ws[ntile(512)][k(8)][c(16)][lane(32)][2]
// so the main kernel's B load is one coalesced global_load_b64 per (k,c).
// One float2 per thread; gather-read, coalesced-write. 16 MB each way.
// ---------------------------------------------------------------------------
__global__ __launch_bounds__(256) void prep_weights_kernel(
    const float* __restrict__ w, float* __restrict__ ws)
{
    const int t = blockIdx.x * 256 + threadIdx.x;   // [0, 512*8*16*32)
    const int lane  = t & 31;
    const int c     = (t >> 5) & 15;
    const int k     = (t >> 9) & 7;
    const int ntile = t >> 12;
    const int l15 = lane & 15, hi = lane >> 4;
    const int n    = ntile * 16 + l15;
    const int kidx = k * IPB + c * 4 + 2 * hi;
    v2f a = *reinterpret_cast<const v2f*>(w + (size_t)n * TOTAL_INPUTS + kidx);
    a.x = __builtin_amdgcn_fmed3f(a.x, 0.0f, __builtin_inff());
    a.y = __builtin_amdgcn_fmed3f(a.y, 0.0f, __builtin_inff());
    *reinterpret_cast<v2f*>(ws + (size_t)t * 2) = a;
}

// ---------------------------------------------------------------------------
// Main kernel. One wave = 16 (batch) x 16 (neuron) tile, 8 branch accumulators.
// PREPPED: B fragments from pre-swizzled d_ws; else strided loads + med3 clamp.
// ---------------------------------------------------------------------------
template <bool PREPPED>
__global__ __launch_bounds__(256) void dendritic_kernel(
    const float* __restrict__ inputs,    // (1024, 512)
    const float* __restrict__ weights,   // raw (8192,8,64) or prepped d_ws
    const float* __restrict__ g_syn,     // (1024, 8192, 8)
    const float* __restrict__ plateaus,  // (1024, 8192, 8)
    const float* __restrict__ g_e,       // (1024, 8192)
    const float* __restrict__ v_mem,     // (1024, 8192)
    float* __restrict__ out_spikes,      // (1024, 8192)
    float* __restrict__ out_v)           // (1024, 8192)
{
    __shared__ __align__(16) float xs[16 * XS_STRIDE];

    const int tid    = threadIdx.x;
    const int bbase  = blockIdx.y * 16;    // batch tile
    const int nbase0 = blockIdx.x * 128;   // neuron tile (8 waves x 16)

    // ---- stage the 16x512 input tile into LDS (2048 float4, 8 per thread) ----
    for (int t = tid; t < 16 * 128; t += 256) {
        const int row = t >> 7;      // 128 float4 per row
        const int c4  = t & 127;
        const float4 v = *reinterpret_cast<const float4*>(
            inputs + (size_t)(bbase + row) * TOTAL_INPUTS + c4 * 4);
        *reinterpret_cast<float4*>(&xs[row * XS_STRIDE + c4 * 4]) = v;
    }
    __syncthreads();

    const int wave = tid >> 5;
    const int lane = tid & 31;
    const int l15  = lane & 15;
    const int hi   = lane >> 4;          // 0: K, K+1  |  1: K+2, K+3
    const int ncol = nbase0 + wave * 16 + l15;

    // per-branch 16x16 fp32 accumulators
    v8f acc[N_BRANCHES];
#pragma unroll
    for (int k = 0; k < N_BRANCHES; ++k) acc[k] = (v8f){};

    // B-fragment base pointers
    const float* wrow  = weights + (size_t)ncol * TOTAL_INPUTS;                 // raw path
    const float* wfrag = weights + (((size_t)blockIdx.x * 8 + wave) * 8192)    // prepped path:
                                 + lane * 2;                                    // ntile*8*16*32*2

    // ---- 8 branches x 16 chunks of K=4 -> 128 v_wmma_f32_16x16x4_f32 ----
#pragma unroll
    for (int k = 0; k < N_BRANCHES; ++k) {
#pragma unroll
        for (int c = 0; c < 16; ++c) {
            const int kidx = k * IPB + c * 4 + 2 * hi;
            // A fragment from LDS: X[bbase+l15][kidx .. kidx+1]
            v2f a = *reinterpret_cast<const v2f*>(&xs[l15 * XS_STRIDE + kidx]);
            v2f bw;
            if (PREPPED) {
                // coalesced: 32 lanes read 256 contiguous bytes
                bw = *reinterpret_cast<const v2f*>(wfrag + (k * 16 + c) * 64);
            } else {
                bw = *reinterpret_cast<const v2f*>(wrow + kidx);
                bw.x = __builtin_amdgcn_fmed3f(bw.x, 0.0f, __builtin_inff());
                bw.y = __builtin_amdgcn_fmed3f(bw.y, 0.0f, __builtin_inff());
            }
            acc[k] = __builtin_amdgcn_wmma_f32_16x16x4_f32(
                false, a, false, bw, (short)0, acc[k], false, false);
        }
    }

    // ---- epilogue: per accumulator row r -> element (M = r + 8*hi, N = l15) ----
#pragma unroll
    for (int r = 0; r < 8; ++r) {
        const int m = r + 8 * hi;
        const size_t idx = (size_t)(bbase + m) * N_NEURONS + ncol;

        // streaming state: non-temporal so it doesn't evict W/X from L2
        const v4f g0 = __builtin_nontemporal_load(
            reinterpret_cast<const v4f*>(g_syn + idx * 8));
        const v4f g1 = __builtin_nontemporal_load(
            reinterpret_cast<const v4f*>(g_syn + idx * 8 + 4));
        const v4f p0 = __builtin_nontemporal_load(
            reinterpret_cast<const v4f*>(plateaus + idx * 8));
        const v4f p1 = __builtin_nontemporal_load(
            reinterpret_cast<const v4f*>(plateaus + idx * 8 + 4));
        const float gsv[8] = {g0.x, g0.y, g0.z, g0.w, g1.x, g1.y, g1.z, g1.w};
        const float plv[8] = {p0.x, p0.y, p0.z, p0.w, p1.x, p1.y, p1.z, p1.w};

        // sum_k 2*tanh(tot_k/2) = 16 - 4 * sum_k rcp(exp(tot_k)+1)
        float rsum = 0.0f;
#pragma unroll
        for (int k = 0; k < N_BRANCHES; ++k) {
            const float wgt = acc[k][r];
            const float g   = SYN_DECAY * gsv[k] + wgt;
            const bool supra = (g > 0.3f);
            const float nm  = g * (supra ? 3.0f : 0.8f);
            float p = PLATEAU_DECAY * plv[k];
            if (supra) p = fmaxf(p, nm);
            const float tot = nm + p;
            const float e   = __builtin_amdgcn_exp2f(tot * LOG2E);
            rsum += __builtin_amdgcn_rcpf(e + 1.0f);
        }
        const float soma = 16.0f - 4.0f * rsum;

        const float gev = __builtin_nontemporal_load(g_e + idx);
        const float vm  = __builtin_nontemporal_load(v_mem + idx);
        const float ge  = E_DECAY * gev + soma;
        float v = vm + DT_OVER_TAU * (ge * (3.0f - vm) - vm);
        const float spike = (v >= 1.0f) ? 1.0f : 0.0f;
        v = (v >= 1.0f) ? 0.0f : v;

        __builtin_nontemporal_store(spike, out_spikes + idx);
        __builtin_nontemporal_store(v, out_v + idx);
    }
}

extern "C" void kernel_launch(void* const* d_in, const int* in_sizes, int n_in,
                              void* d_out, int out_size, void* d_ws, size_t ws_size,
                              hipStream_t stream) {
    const float* inputs   = (const float*)d_in[0];
    const float* weights  = (const float*)d_in[1];
    const float* g_syn    = (const float*)d_in[2];
    const float* plateaus = (const float*)d_in[3];
    const float* g_e      = (const float*)d_in[4];
    const float* v_mem    = (const float*)d_in[5];

    float* out        = (float*)d_out;
    float* out_spikes = out;                                   // first output
    float* out_v      = out + (size_t)BATCH * N_NEURONS;       // second output

    const size_t W_ELEMS = (size_t)N_NEURONS * TOTAL_INPUTS;   // 4M floats
    dim3 grid(N_NEURONS / 128, BATCH / 16);

    if (ws_size >= W_ELEMS * sizeof(float)) {
        float* wprep = (float*)d_ws;
        // 2M float2 elements, 256 threads each handling one
        prep_weights_kernel<<<(W_ELEMS / 2) / 256, 256, 0, stream>>>(weights, wprep);
        dendritic_kernel<true><<<grid, 256, 0, stream>>>(
            inputs, wprep, g_syn, plateaus, g_e, v_mem, out_spikes, out_v);
    } else {
        dendritic_kernel<false><<<grid, 256, 0, stream>>>(
            inputs, weights, g_syn, plateaus, g_e, v_mem, out_spikes, out_v);
    }
}